// weisfeiler_lehman_conv_19688130084889
// MI455X (gfx1250) — compile-verified
//
#include <hip/hip_runtime.h>

// Problem constants (match reference)
#define B_  4
#define N_  4096
#define KW  16     // label width == WMMA tile N
#define C_  8
#define T_  3
#define WPB 8      // waves per block (1 wave = one 16-row output tile)

typedef __attribute__((ext_vector_type(2))) float v2f;
typedef __attribute__((ext_vector_type(8))) float v8f;

// ---------------------------------------------------------------------------
// Kernel 1: threshold + bit-pack adjacency: bits[b][n][w] bit L = (adj >= 1)
// 256 MB int32 -> 8 MB bitmask (L2-resident for the matmul sweeps).
// ---------------------------------------------------------------------------
__global__ __launch_bounds__(256)
void wl_pack(const int* __restrict__ adj, unsigned* __restrict__ bits) {
    size_t idx = (size_t)blockIdx.x * 256 + threadIdx.x;   // over B*N*N
    bool pred = adj[idx] >= 1;
#if __has_builtin(__builtin_amdgcn_ballot_w32)
    unsigned w = __builtin_amdgcn_ballot_w32(pred);
#else
    unsigned w = (unsigned)__ballot(pred);
#endif
    if ((threadIdx.x & 31) == 0) bits[idx >> 5] = w;
}

// ---------------------------------------------------------------------------
// Kernel 2: Xout = mask @ Xin   ([N x N] 0/1  x  [N x 16]) per batch,
// using V_WMMA_F32_16X16X4_F32 (fp32 matrix pipe, exact vs reference).
// One wave computes one 16x16 output tile (16 rows x full K=16).
// ---------------------------------------------------------------------------
__global__ __launch_bounds__(WPB * 32)
void wl_spmm_pow(const unsigned* __restrict__ maskbits,
                 const float* __restrict__ Xin,
                 float* __restrict__ Xout) {
    const int lane    = threadIdx.x & 31;
    const int wave    = threadIdx.x >> 5;
    const int b       = blockIdx.y;
    const int rowTile = (blockIdx.x * WPB + wave) * 16;
    const int m       = lane & 15;          // A-row / B-col / D-col for this lane
    const int koff    = (lane >> 4) << 1;   // K sub-offset: 0 (lanes 0-15), 2 (lanes 16-31)

    const unsigned* Arow = maskbits + (size_t)b * N_ * (N_ / 32)
                                    + (size_t)(rowTile + m) * (N_ / 32);
    const float*    X    = Xin  + (size_t)b * N_ * KW;
    float*          O    = Xout + (size_t)b * N_ * KW;

    v8f acc = {};
    for (int s = 0; s < N_; s += 32) {
        const unsigned w = Arow[s >> 5];    // 32 adjacency bits for this lane's row
#pragma unroll
        for (int kk = 0; kk < 32; kk += 4) {
            v2f a, bb;
            // A-frag: 16x4 f32, lane holds row m, K = kk+koff, kk+koff+1
            a.x = (float)((w >> (kk + koff)) & 1u);
            a.y = (float)((w >> (kk + koff + 1)) & 1u);
            // B-frag: 4x16 f32, lane holds col m, same K split
            const float* xb = X + (size_t)(s + kk + koff) * KW + m;
            bb.x = xb[0];
            bb.y = xb[KW];
            acc = __builtin_amdgcn_wmma_f32_16x16x4_f32(
                false, a, false, bb, (short)0, acc, false, false);
        }
    }

    // D layout: VGPR r -> row rowTile + r + 8*(lane>=16), col m
    const int orow = rowTile + ((lane >> 4) << 3);
#pragma unroll
    for (int r = 0; r < 8; ++r)
        O[(size_t)(orow + r) * KW + m] = acc[r];
}

// ---------------------------------------------------------------------------
// Kernel 3: out[b,c,n,k] = L + e1*P1 + e2*P2 + e3*P3, with e_j the elementary
// symmetric polynomials of kernels[c,0..2,k] (diagonal scalings commute).
// ---------------------------------------------------------------------------
__global__ __launch_bounds__(256)
void wl_combine(const float* __restrict__ labels,
                const float* __restrict__ kern,
                const float* __restrict__ P1,
                const float* __restrict__ P2,
                const float* __restrict__ P3,
                float* __restrict__ out) {
    int idx = blockIdx.x * 256 + threadIdx.x;   // exactly B*C*N*KW threads
    int k = idx & (KW - 1);
    int n = (idx >> 4) & (N_ - 1);
    int c = (idx >> 16) & (C_ - 1);
    int b = idx >> 19;

    float k0 = kern[(c * T_ + 0) * KW + k];
    float k1 = kern[(c * T_ + 1) * KW + k];
    float k2 = kern[(c * T_ + 2) * KW + k];
    float e1 = k0 + k1 + k2;
    float e2 = k0 * k1 + k0 * k2 + k1 * k2;
    float e3 = k0 * k1 * k2;

    size_t p = ((size_t)b * N_ + n) * KW + k;
    out[idx] = labels[p] + e1 * P1[p] + e2 * P2[p] + e3 * P3[p];
}

// ---------------------------------------------------------------------------
extern "C" void kernel_launch(void* const* d_in, const int* in_sizes, int n_in,
                              void* d_out, int out_size, void* d_ws, size_t ws_size,
                              hipStream_t stream) {
    const float* labels = (const float*)d_in[0];   // [B,N,K] f32
    const int*   adj    = (const int*)d_in[1];     // [B,N,N] i32
    const float* kern   = (const float*)d_in[2];   // [C,T,K] f32
    float*       out    = (float*)d_out;           // [B,C,N,K] f32

    // Workspace layout: bitmask (8 MB) then P1,P2,P3 (1 MB each)
    unsigned* bits = (unsigned*)d_ws;
    float* P1 = (float*)((char*)d_ws + (size_t)B_ * N_ * (N_ / 8));
    float* P2 = P1 + (size_t)B_ * N_ * KW;
    float* P3 = P2 + (size_t)B_ * N_ * KW;

    // 1) threshold + bit-pack adjacency
    {
        long long total = (long long)B_ * N_ * N_;
        wl_pack<<<(unsigned)(total / 256), 256, 0, stream>>>(adj, bits);
    }

    // 2) P1 = A L ; P2 = A P1 ; P3 = A P2   (fp32 WMMA)
    dim3 grid(N_ / 16 / WPB, B_);
    dim3 blk(WPB * 32);
    wl_spmm_pow<<<grid, blk, 0, stream>>>(bits, labels, P1);
    wl_spmm_pow<<<grid, blk, 0, stream>>>(bits, P1, P2);
    wl_spmm_pow<<<grid, blk, 0, stream>>>(bits, P2, P3);

    // 3) per-channel combine via elementary symmetric polynomials
    {
        int total = B_ * C_ * N_ * KW;   // 2,097,152
        wl_combine<<<total / 256, 256, 0, stream>>>(labels, kern, P1, P2, P3, out);
    }
}